// SimpleGCNLayer_57526791962769
// MI455X (gfx1250) — compile-verified
//
#include <hip/hip_runtime.h>
#include <hip/hip_bf16.h>
#include <math.h>

// Problem constants (fixed by the reference's setup_inputs)
#define Bsz 4
#define Nn  2048
#define Dd  128
#define De  8

typedef float v2f __attribute__((ext_vector_type(2)));
typedef float v4f __attribute__((ext_vector_type(4)));
typedef float v8f __attribute__((ext_vector_type(8)));

// ---------------------------------------------------------------------------
// Kernel 0: transpose W (128x128) into WT so kernel 3's loads are coalesced.
// ---------------------------------------------------------------------------
__global__ void k0_transposeW(const float* __restrict__ W, float* __restrict__ WT) {
    int idx = blockIdx.x * 256 + threadIdx.x;     // 0..16383
    int d = idx >> 7;
    int k = idx & 127;
    WT[k * Dd + d] = W[idx];                      // WT[k][d] = W[d][k]
}

// ---------------------------------------------------------------------------
// Kernel 1 (bandwidth-critical): one block per (b,i) row.
//  - stream e row (64KB) with NT loads, copy to out_e with NT stores
//  - scores = L2 norm over De=8, softmax over j (2048) in registers+LDS
//  - write w row (temporal -> stays in L2 for the GEMM)
// ---------------------------------------------------------------------------
__global__ __launch_bounds__(256) void k1_scores_softmax(
        const float* __restrict__ e, float* __restrict__ e_out,
        float* __restrict__ wmat) {
    const int row = blockIdx.x;                   // b*N + i
    const size_t base = (size_t)row * Nn * De;    // float offset of e row
    const v4f* ein  = reinterpret_cast<const v4f*>(e + base);
    v4f* eout       = reinterpret_cast<v4f*>(e_out + base);
    float* wrow     = wmat + (size_t)row * Nn;
    const int t = threadIdx.x;

    float s[8];
    float lmax = -3.402823466e38f;
    #pragma unroll
    for (int q = 0; q < 8; ++q) {
        int j = t + q * 256;
        v4f v0 = __builtin_nontemporal_load(&ein[j * 2 + 0]);
        v4f v1 = __builtin_nontemporal_load(&ein[j * 2 + 1]);
        __builtin_nontemporal_store(v0, &eout[j * 2 + 0]);
        __builtin_nontemporal_store(v1, &eout[j * 2 + 1]);
        float ss = v0.x * v0.x + v0.y * v0.y + v0.z * v0.z + v0.w * v0.w
                 + v1.x * v1.x + v1.y * v1.y + v1.z * v1.z + v1.w * v1.w;
        s[q] = sqrtf(ss);
        lmax = fmaxf(lmax, s[q]);
    }

    __shared__ float red[256];
    red[t] = lmax;
    __syncthreads();
    for (int sd = 128; sd > 0; sd >>= 1) {
        if (t < sd) red[t] = fmaxf(red[t], red[t + sd]);
        __syncthreads();
    }
    const float rmax = red[0];
    __syncthreads();

    float lsum = 0.0f;
    #pragma unroll
    for (int q = 0; q < 8; ++q) {
        s[q] = __expf(s[q] - rmax);
        lsum += s[q];
    }
    red[t] = lsum;
    __syncthreads();
    for (int sd = 128; sd > 0; sd >>= 1) {
        if (t < sd) red[t] += red[t + sd];
        __syncthreads();
    }
    const float inv = 1.0f / red[0];
    #pragma unroll
    for (int q = 0; q < 8; ++q)
        wrow[t + q * 256] = s[q] * inv;           // temporal: keep in L2
}

// ---------------------------------------------------------------------------
// Kernel 2: agg[b] = w[b] (2048x2048) @ h[b] (2048x128) via f32 WMMA 16x16x4.
// Block = 256 threads (8 waves): 16-row x 128-col stripe; wave wv owns a
// 16x16 tile at cols [16*wv, 16*wv+16). K staged in LDS in chunks of 64.
// A LDS stride 68 (banks 4m / 4m+2 -> conflict-free); H stride 136
// (lane halves offset by 16 banks -> conflict-free).
// ---------------------------------------------------------------------------
__global__ __launch_bounds__(256) void k2_agg_wmma(
        const float* __restrict__ wmat, const float* __restrict__ h,
        float* __restrict__ agg) {
    const int blk  = blockIdx.x;                  // 0 .. B*(N/16)-1
    const int b    = blk / (Nn / 16);
    const int ti   = blk % (Nn / 16);
    const int row0 = ti * 16;
    const float* A = wmat + (size_t)b * Nn * Nn;
    const float* H = h    + (size_t)b * Nn * Dd;
    float*       C = agg  + (size_t)b * Nn * Dd;

    __shared__ float As[16 * 68];                 //  4352 B
    __shared__ float Hs[64 * 136];                // 34816 B

    const int t     = threadIdx.x;
    const int wv    = t >> 5;                     // wave 0..7
    const int lane  = t & 31;
    const int col0  = wv * 16;
    const int m     = lane & 15;
    const int kh2   = (lane >> 4) * 2;            // 0 or 2 (K half select)

    v8f acc = {};

    for (int k0 = 0; k0 < Nn; k0 += 64) {
        __syncthreads();
        // A tile 16x64 -> LDS (coalesced: 64 consecutive floats per row)
        #pragma unroll
        for (int q = 0; q < 4; ++q) {
            int idx = t + q * 256;                // 0..1023
            int r = idx >> 6, c = idx & 63;
            As[r * 68 + c] = A[(size_t)(row0 + r) * Nn + k0 + c];
        }
        // H tile 64x128 -> LDS via float4 (2048 float4, 8 per thread)
        #pragma unroll
        for (int q = 0; q < 8; ++q) {
            int idx = t + q * 256;                // 0..2047
            int r = idx >> 5, c4 = idx & 31;
            v4f v = *reinterpret_cast<const v4f*>(&H[(size_t)(k0 + r) * Dd + c4 * 4]);
            float* dst = &Hs[r * 136 + c4 * 4];
            dst[0] = v.x; dst[1] = v.y; dst[2] = v.z; dst[3] = v.w;
        }
        __syncthreads();

        #pragma unroll
        for (int kk = 0; kk < 64; kk += 4) {
            v2f af, bf;
            af.x = As[m * 68 + kk + kh2 + 0];     // A[m][kk + kh2 + {0,1}]
            af.y = As[m * 68 + kk + kh2 + 1];
            bf.x = Hs[(kk + kh2 + 0) * 136 + col0 + m];  // B[k][n], n = col0+m
            bf.y = Hs[(kk + kh2 + 1) * 136 + col0 + m];
            acc = __builtin_amdgcn_wmma_f32_16x16x4_f32(
                      false, af, false, bf, (short)0, acc, false, false);
        }
    }

    // C/D layout: VGPR v -> M = v (lanes 0-15) or v+8 (lanes 16-31); col = lane%16
    const int mofs = (lane >= 16) ? 8 : 0;
    #pragma unroll
    for (int v = 0; v < 8; ++v)
        C[(size_t)(row0 + v + mofs) * Dd + col0 + m] = acc[v];
}

// ---------------------------------------------------------------------------
// Kernel 3: z = (h+agg) @ W^T + b ; out = relu(LN(z)) + h.
// 128 threads (thread t = output dim d), 16 rows per block.
// WT[k*128+d] loads are lane-coalesced; LDS tree reduction for mean/var.
// ---------------------------------------------------------------------------
__global__ __launch_bounds__(128) void k3_linear_ln(
        const float* __restrict__ h, const float* __restrict__ agg,
        const float* __restrict__ WT, const float* __restrict__ Wb,
        const float* __restrict__ ln_g, const float* __restrict__ ln_b,
        float* __restrict__ out) {
    const int t = threadIdx.x;                    // output dim d
    const int row0 = blockIdx.x * 16;

    __shared__ float xs[128];
    __shared__ float red[128];
    __shared__ float red2[128];

    const float wb = Wb[t];
    const float gg = ln_g[t];
    const float bb = ln_b[t];

    for (int r = 0; r < 16; ++r) {
        const size_t row = (size_t)(row0 + r);
        const float hval = h[row * Dd + t];
        __syncthreads();
        xs[t] = hval + agg[row * Dd + t];
        __syncthreads();

        float z = wb;
        #pragma unroll 8
        for (int k = 0; k < 128; ++k)
            z = fmaf(xs[k], WT[k * Dd + t], z);

        red[t]  = z;
        red2[t] = z * z;
        __syncthreads();
        for (int sd = 64; sd > 0; sd >>= 1) {
            if (t < sd) { red[t] += red[t + sd]; red2[t] += red2[t + sd]; }
            __syncthreads();
        }
        const float mu  = red[0] * (1.0f / 128.0f);
        const float var = red2[0] * (1.0f / 128.0f) - mu * mu;
        float y = (z - mu) * rsqrtf(var + 1e-5f) * gg + bb;
        y = fmaxf(y, 0.0f) + hval;
        out[row * Dd + t] = y;
    }
}

// ---------------------------------------------------------------------------
extern "C" void kernel_launch(void* const* d_in, const int* in_sizes, int n_in,
                              void* d_out, int out_size, void* d_ws, size_t ws_size,
                              hipStream_t stream) {
    const float* h    = (const float*)d_in[0];   // [B,N,D]
    const float* e    = (const float*)d_in[1];   // [B,N,N,De]
    const float* W_w  = (const float*)d_in[2];   // [D,D]
    const float* W_b  = (const float*)d_in[3];   // [D]
    const float* ln_g = (const float*)d_in[4];   // [D]
    const float* ln_b = (const float*)d_in[5];   // [D]

    float* out_h = (float*)d_out;                              // B*N*D
    float* out_e = out_h + (size_t)Bsz * Nn * Dd;              // B*N*N*De

    float* ws   = (float*)d_ws;
    float* wmat = ws;                                          // B*N*N
    float* agg  = wmat + (size_t)Bsz * Nn * Nn;                // B*N*D
    float* WT   = agg  + (size_t)Bsz * Nn * Dd;                // D*D

    k0_transposeW<<<(Dd * Dd) / 256, 256, 0, stream>>>(W_w, WT);
    k1_scores_softmax<<<Bsz * Nn, 256, 0, stream>>>(e, out_e, wmat);
    k2_agg_wmma<<<Bsz * (Nn / 16), 256, 0, stream>>>(wmat, h, agg);
    k3_linear_ln<<<(Bsz * Nn) / 16, 128, 0, stream>>>(h, agg, WT, W_b, ln_g, ln_b, out_h);
}